// MambaConv1D_60078002536790
// MI455X (gfx1250) — compile-verified
//
#include <hip/hip_runtime.h>

// Problem constants (from the reference)
#define BATCH   4
#define CHANS   4096
#define L_IN    4096
#define L_OUT   4099   // L_IN + 2*PAD - K + 1 = 4096 + 6 - 4 + 1
#define KSIZE   4
#define PADL    3

// LDS row buffer: x placed at float index 4 (16B aligned dest for async b128),
// so smem[j+1] == x_padded[j] == x[j-3]. Output i reads smem[i+1 .. i+4].
// Max read index = 4098 + 4 = 4102; allocate 4104 (16B multiple).
#define SMEM_FLOATS 4104

__global__ __launch_bounds__(256) void MambaConv1D_dwconv_kernel(
    const float* __restrict__ x,     // [B*C, L_IN]
    const float* __restrict__ w,     // [C, K]  (kernel[c,0,k])
    const float* __restrict__ bias,  // [C]
    float* __restrict__ y)           // [B*C, L_OUT]
{
    __shared__ __attribute__((aligned(16))) float smem[SMEM_FLOATS];

    const int tid = threadIdx.x;
    const int row = blockIdx.x;              // 0 .. B*C-1
    const int c   = row & (CHANS - 1);       // channel (C is a power of 2)

    const float* xrow = x + (size_t)row * L_IN;
    float*       yrow = y + (size_t)row * L_OUT;

    // Zero the halos: smem[0..3] (left, x[-4..-1]) and smem[4100..4103] (right).
    if (tid < 4)              smem[tid] = 0.0f;
    if (tid >= 4 && tid < 8)  smem[4096 + tid] = 0.0f;   // 4100..4103

    // ---- CDNA5 async global->LDS staging (ASYNCcnt-tracked) ----
    // 4 rounds x 256 lanes x 16B = 16 KB = the full row into smem[4..4099].
    #pragma unroll
    for (int it = 0; it < 4; ++it) {
        const int f = it * 1024 + tid * 4;                       // float idx in row
        const unsigned lds_off = (unsigned)(size_t)(&smem[4 + f]); // LDS byte offset (low 32 bits of flat ptr)
        const float*   gaddr   = xrow + f;                        // 16B aligned
        asm volatile("global_load_async_to_lds_b128 %0, %1, off"
                     :: "v"(lds_off), "v"(gaddr)
                     : "memory");
    }
    asm volatile("s_wait_asynccnt 0x0" ::: "memory");  // this wave's async copies done
    __syncthreads();                                   // all waves' copies visible

    // Block-uniform taps + bias -> scalar loads (w row is 16B aligned: K=4 f32).
    const float4 wv = *(const float4*)(w + (size_t)c * KSIZE);
    const float  bb = bias[c];

    // ---- Compute: 4 outputs/thread/chunk from aligned LDS b128 reads ----
    // out[i] = w0*smem[i+1] + w1*smem[i+2] + w2*smem[i+3] + w3*smem[i+4] + bias
    #pragma unroll
    for (int ch = 0; ch < 4; ++ch) {
        const int base = ch * 1024 + tid * 4;          // 0..4092, covers outputs 0..4095
        const float4 p0 = *(const float4*)(&smem[base]);      // smem[base..base+3]
        const float4 p1 = *(const float4*)(&smem[base + 4]);  // smem[base+4..base+7]

        const float o0 = wv.x*p0.y + wv.y*p0.z + wv.z*p0.w + wv.w*p1.x + bb;
        const float o1 = wv.x*p0.z + wv.y*p0.w + wv.z*p1.x + wv.w*p1.y + bb;
        const float o2 = wv.x*p0.w + wv.y*p1.x + wv.z*p1.y + wv.w*p1.z + bb;
        const float o3 = wv.x*p1.x + wv.y*p1.y + wv.z*p1.z + wv.w*p1.w + bb;

        yrow[base + 0] = o0;
        yrow[base + 1] = o1;
        yrow[base + 2] = o2;
        yrow[base + 3] = o3;
    }

    // Tail outputs 4096..4098 (right-padded region).
    if (tid < 3) {
        const int i = 4096 + tid;
        yrow[i] = wv.x*smem[i+1] + wv.y*smem[i+2] + wv.z*smem[i+3] + wv.w*smem[i+4] + bb;
    }
}

extern "C" void kernel_launch(void* const* d_in, const int* in_sizes, int n_in,
                              void* d_out, int out_size, void* d_ws, size_t ws_size,
                              hipStream_t stream) {
    (void)in_sizes; (void)n_in; (void)out_size; (void)d_ws; (void)ws_size;
    const float* x    = (const float*)d_in[0];   // (B, C, L) f32
    const float* w    = (const float*)d_in[1];   // (C, 1, K) f32
    const float* bias = (const float*)d_in[2];   // (C,) f32
    float*       y    = (float*)d_out;           // (B, C, L_OUT) f32

    const int rows = BATCH * CHANS;              // 16384 blocks, one row each
    MambaConv1D_dwconv_kernel<<<dim3(rows), dim3(256), 0, stream>>>(x, w, bias, y);
}